// PCNet_51204600103671
// MI455X (gfx1250) — compile-verified
//
#include <hip/hip_runtime.h>

typedef __attribute__((ext_vector_type(2))) float v2f;
typedef __attribute__((ext_vector_type(8))) float v8f;

#define B_ROWS 131072
#define IN_DIM 16
#define H_DIM 64
#define OUT_DIM 16
#define STEPS 30
#define WPB 8              // waves per block
#define ROWS_PER_WAVE 16

__device__ __forceinline__ v8f wmma4(v2f a, v2f b, v8f c) {
  // D(16x16,f32) = A(16x4,f32) x B(4x16,f32) + C
  return __builtin_amdgcn_wmma_f32_16x16x4_f32(false, a, false, b, (short)0, c, false, false);
}

__global__ __launch_bounds__(32 * WPB)
void pcnet_kernel(const float* __restrict__ x,  const float* __restrict__ W1,
                  const float* __restrict__ b1, const float* __restrict__ W2,
                  const float* __restrict__ b2, const float* __restrict__ x1i,
                  const float* __restrict__ x2i, float* __restrict__ out)
{
  __shared__ float sh[WPB * H_DIM * 16];           // 32 KB: one 64x16 f32 slab per wave
  const int wv   = threadIdx.x >> 5;
  const int lane = threadIdx.x & 31;
  const int l15  = lane & 15;
  const int half = (lane >> 4) & 1;
  const int row  = (blockIdx.x * WPB + wv) * ROWS_PER_WAVE;
  float* L = sh + wv * (H_DIM * 16);

  // ---- constant W2 in A-layout: M=out(lane), K=hidden (16 chunks of 4) ----
  v2f w2a[16];
  #pragma unroll
  for (int k = 0; k < 16; ++k)
    w2a[k] = *(const v2f*)(W2 + l15 * H_DIM + 4 * k + 2 * half);

  // ---- constant W2^T in A-layout: 4 M-tiles of hidden, K=out (4 chunks) ----
  v2f w2t[4][4];
  #pragma unroll
  for (int t = 0; t < 4; ++t)
    #pragma unroll
    for (int k = 0; k < 4; ++k) {
      v2f v;
      v.x = W2[(4 * k + 0 + 2 * half) * H_DIM + 16 * t + l15];
      v.y = W2[(4 * k + 1 + 2 * half) * H_DIM + 16 * t + l15];
      w2t[t][k] = v;
    }

  v8f b2cd = *(const v8f*)(b2 + 8 * half);         // C/D layout [out x batch] bias

  // ---- mu1 = sigmoid(x @ W1^T + b1), computed as D[h][m] = W1 x x^T ----
  v2f xb[4];                                       // B-layout x^T: K=in, N=batch
  #pragma unroll
  for (int k = 0; k < 4; ++k)
    xb[k] = *(const v2f*)(x + (row + l15) * IN_DIM + 4 * k + 2 * half);

  #pragma unroll
  for (int t = 0; t < 4; ++t) {
    v2f w1a[4];                                    // A-layout W1 tile: M=hidden, K=in
    #pragma unroll
    for (int k = 0; k < 4; ++k)
      w1a[k] = *(const v2f*)(W1 + (16 * t + l15) * IN_DIM + 4 * k + 2 * half);
    v8f acc;
    #pragma unroll
    for (int i = 0; i < 8; ++i) acc[i] = b1[16 * t + i + 8 * half];
    #pragma unroll
    for (int k = 0; k < 4; ++k) acc = wmma4(w1a[k], xb[k], acc);
    #pragma unroll
    for (int i = 0; i < 8; ++i) {                  // sigmoid, stage into LDS [m*64+h]
      float e  = __builtin_amdgcn_exp2f(-acc[i] * 1.44269504088896340736f);
      float mu = __builtin_amdgcn_rcpf(1.0f + e);
      L[l15 * H_DIM + 16 * t + i + 8 * half] = mu;
    }
  }
  __syncthreads();
  v2f mu1b[16];                                    // mu1^T in B-layout (K=hidden chunks)
  #pragma unroll
  for (int k = 0; k < 16; ++k)
    mu1b[k] = *(const v2f*)(L + l15 * H_DIM + 4 * k + 2 * half);

  // ---- state: x1 in B-layout, x2 in C/D [out x batch] ----
  v2f x1b[16];
  #pragma unroll
  for (int k = 0; k < 16; ++k)
    x1b[k] = *(const v2f*)(x1i + (row + l15) * H_DIM + 4 * k + 2 * half);
  v8f x2cd = *(const v8f*)(x2i + (row + l15) * OUT_DIM + 8 * half);

  // ---- 30 inference steps, fully resident ----
  for (int s = 0; s < STEPS; ++s) {
    // mu2pre^T = W2 @ x1^T + b2   (D: [out x batch])
    v8f m2 = b2cd;
    #pragma unroll
    for (int k = 0; k < 16; ++k) m2 = wmma4(w2a[k], x1b[k], m2);

    v8f mu2;
    #pragma unroll
    for (int i = 0; i < 8; ++i) {
      float e = __builtin_amdgcn_exp2f(-m2[i] * 1.44269504088896340736f);
      mu2[i]  = __builtin_amdgcn_rcpf(1.0f + e);
    }
    v8f e2 = x2cd - mu2;
    x2cd   = x2cd - 0.2f * e2;                     // x2 -= LR*2*e2
    v8f h2 = e2 * mu2 * (1.0f - mu2);

    // stage h2 [out x batch] -> B-layout h2^T (K=out, N=batch)
    #pragma unroll
    for (int i = 0; i < 8; ++i)
      L[l15 * OUT_DIM + i + 8 * half] = h2[i];
    __syncthreads();
    v2f h2b[4];
    #pragma unroll
    for (int k = 0; k < 4; ++k)
      h2b[k] = *(const v2f*)(L + l15 * OUT_DIM + 4 * k + 2 * half);

    // corr^T = W2^T @ h2^T  -> 4 D tiles [hidden x batch] -> LDS [m*64+h]
    #pragma unroll
    for (int t = 0; t < 4; ++t) {
      v8f ct = {};
      #pragma unroll
      for (int k = 0; k < 4; ++k) ct = wmma4(w2t[t][k], h2b[k], ct);
      #pragma unroll
      for (int i = 0; i < 8; ++i)
        L[l15 * H_DIM + 16 * t + i + 8 * half] = ct[i];
    }
    __syncthreads();
    // x1 = 0.8*x1 + 0.2*mu1 + 0.2*corr  (all in B-layout)
    #pragma unroll
    for (int k = 0; k < 16; ++k) {
      v2f c = *(const v2f*)(L + l15 * H_DIM + 4 * k + 2 * half);
      x1b[k] = 0.8f * x1b[k] + 0.2f * mu1b[k] + 0.2f * c;
    }
  }

  // ---- outputs: x1 flat [B,64] then x2 flat [B,16] ----
  #pragma unroll
  for (int k = 0; k < 16; ++k)
    *(v2f*)(out + (size_t)(row + l15) * H_DIM + 4 * k + 2 * half) = x1b[k];
  *(v8f*)(out + (size_t)B_ROWS * H_DIM + (size_t)(row + l15) * OUT_DIM + 8 * half) = x2cd;
}

extern "C" void kernel_launch(void* const* d_in, const int* in_sizes, int n_in,
                              void* d_out, int out_size, void* d_ws, size_t ws_size,
                              hipStream_t stream) {
  const float* x   = (const float*)d_in[0];
  const float* W1  = (const float*)d_in[1];
  const float* b1  = (const float*)d_in[2];
  const float* W2  = (const float*)d_in[3];
  const float* b2  = (const float*)d_in[4];
  const float* x1i = (const float*)d_in[5];
  const float* x2i = (const float*)d_in[6];
  float* out = (float*)d_out;
  const int blocks = B_ROWS / (WPB * ROWS_PER_WAVE);  // 1024
  pcnet_kernel<<<blocks, 32 * WPB, 0, stream>>>(x, W1, b1, W2, b2, x1i, x2i, out);
}